// VDPReLU_89541478187437
// MI455X (gfx1250) — compile-verified
//
#include <hip/hip_runtime.h>
#include <stdint.h>

typedef __attribute__((ext_vector_type(2))) float v2f;
typedef __attribute__((ext_vector_type(8))) float v8f;

#define MATS      2048        // 32*64 independent 128x128 matrices
#define NDIM      128
#define MAT_ELEMS (NDIM * NDIM)   // 16384
#define MU_ELEMS  (MATS * NDIM)   // 262144

// Convert a generic pointer to a raw LDS byte offset (addrspace(3) ptrtoint).
__device__ __forceinline__ unsigned lds_byte_addr(const void* p) {
  return (unsigned)(unsigned long long)(__attribute__((address_space(3))) const void*)p;
}

__global__ __launch_bounds__(256) void vdp_relu_sigma_kernel(
    const float* __restrict__ mu,
    const float* __restrict__ Sigma,
    float* __restrict__ out) {
  __shared__ float smu[NDIM];

  const int mat = blockIdx.x;            // one 128x128 matrix per workgroup
  const unsigned tid = threadIdx.x;      // 256 threads = 8 wave32
  const float* mu_row = mu + (size_t)mat * NDIM;

  // ---- Stage the 512B mask row into LDS with one wave32 async-to-LDS b128 ----
  if (tid < 32u) {
    unsigned laddr = lds_byte_addr(smu) + (tid << 4);
    unsigned long long gaddr =
        (unsigned long long)(const void*)mu_row + (unsigned long long)(tid << 4);
    asm volatile("global_load_async_to_lds_b128 %0, %1, off"
                 :: "v"(laddr), "v"(gaddr)
                 : "memory");
  }
  asm volatile("s_wait_asynccnt 0" ::: "memory");
  __syncthreads();

  // ---- Fused mu_out = relu(mu) from the LDS-staged row ----
  if (tid < (unsigned)NDIM) {
    float m = smu[tid];
    out[(size_t)mat * NDIM + tid] = m > 0.0f ? m : 0.0f;
  }

  const float* sig_in  = Sigma + (size_t)mat * MAT_ELEMS;
  float*       sig_out = out + MU_ELEMS + (size_t)mat * MAT_ELEMS;

  const unsigned lane   = tid & 31u;
  const unsigned wv     = tid >> 5;          // wave id 0..7 -> tile row i0
  const unsigned i0     = wv << 4;
  const unsigned lanelo = lane & 15u;
  const unsigned rowoff = (lane >> 4) << 3;  // lanes 16..31 hold rows M+8
  const bool     lo     = lane < 16u;

  // Branchless A-fragment mask fetch (all lanes load a valid LDS word).
  const float mvi = smu[i0 + lanelo];
  const float ai  = (lo && mvi > 0.0f) ? 1.0f : 0.0f;
  const v2f   Af  = {ai, 0.0f};

  const unsigned col_lo = lanelo;            // N = lane % 16

#pragma unroll
  for (unsigned jt = 0; jt < 8; ++jt) {
    const unsigned j0 = jt << 4;

    // Branchless B fragment (4x16 f32): only K=0 row populated.
    const float mvj = smu[j0 + lanelo];
    const float bj  = (lo && mvj > 0.0f) ? 1.0f : 0.0f;
    v2f Bf = {bj, 0.0f};
    v8f Cz = {};
    // D[M,N] = mask_i[i0+M] * mask_j[j0+N]  (16x16 tile in one matrix op)
    v8f Mt = __builtin_amdgcn_wmma_f32_16x16x4_f32(
        /*neg_a=*/false, Af, /*neg_b=*/false, Bf,
        /*c_mod=*/(short)0, Cz, /*reuse_a=*/false, /*reuse_b=*/false);

    const unsigned col = j0 + col_lo;

    // Batch all 8 loads first (independent of the WMMA result) so they form
    // a single vmem clause with 8 outstanding NT loads, then mul+store.
    float sv[8];
#pragma unroll
    for (unsigned v = 0; v < 8; ++v) {
      const unsigned row = i0 + rowoff + v;  // D layout: M = v + 8*(lane>=16)
      sv[v] = __builtin_nontemporal_load(sig_in + (size_t)row * NDIM + col);
    }
#pragma unroll
    for (unsigned v = 0; v < 8; ++v) {
      const unsigned row = i0 + rowoff + v;
      __builtin_nontemporal_store(sv[v] * Mt[v],
                                  sig_out + (size_t)row * NDIM + col);
    }
  }
}

extern "C" void kernel_launch(void* const* d_in, const int* in_sizes, int n_in,
                              void* d_out, int out_size, void* d_ws, size_t ws_size,
                              hipStream_t stream) {
  const float* mu    = (const float*)d_in[0];   // (32,64,128) fp32
  const float* Sigma = (const float*)d_in[1];   // (32,64,128,128) fp32
  float* out = (float*)d_out;                   // [mu_out | Sigma_out] fp32

  (void)in_sizes; (void)n_in; (void)out_size; (void)d_ws; (void)ws_size;

  vdp_relu_sigma_kernel<<<MATS, 256, 0, stream>>>(mu, Sigma, out);
}